// PatchedQwen2MoeSparseMoeBlock_80607946211549
// MI455X (gfx1250) — compile-verified
//
#include <hip/hip_runtime.h>
#include <hip/hip_bf16.h>

typedef __attribute__((ext_vector_type(2))) float v2f;
typedef __attribute__((ext_vector_type(8))) float v8f;

#define T_TOK 1024
#define H_DIM 2048
#define E_NUM 32
#define I_DIM 1408
#define IS_DIM 5632
#define CAP 256

#define BM 128
#define BN 64
#define BK 32
#define LDA 36   // padded stride: conflict-free A-fragment column reads
#define LDB 72   // padded stride: half-waves hit disjoint banks on B rows

static __device__ __forceinline__ v8f zero8() {
    v8f z = {0.f, 0.f, 0.f, 0.f, 0.f, 0.f, 0.f, 0.f};
    return z;
}

// Async global -> LDS 128-bit copy (ASYNCcnt tracked, no VGPR staging).
static __device__ __forceinline__ void async_b128(void* lds_dst, const float* gsrc) {
    unsigned loff = (unsigned)(unsigned long long)lds_dst;  // low 32 bits = LDS offset
    asm volatile("global_load_async_to_lds_b128 %0, %1, off"
                 :: "v"(loff), "v"(gsrc) : "memory");
}
static __device__ __forceinline__ void wait_async0() {
    asm volatile("s_wait_asynccnt 0x0" ::: "memory");
}

// ---------------------------------------------------------------------------
// K1: router logits + softmax + top-4 + shared-gate sigmoid. One wave / token.
// ---------------------------------------------------------------------------
__global__ __launch_bounds__(256)
void router_kernel(const float* __restrict__ x, const float* __restrict__ gw,
                   const float* __restrict__ wshg,
                   float* __restrict__ logits, float* __restrict__ topw,
                   int* __restrict__ sel, float* __restrict__ sgate) {
    const int lane = threadIdx.x & 31;
    const int wv = threadIdx.x >> 5;
    const int t = blockIdx.x * 8 + wv;
    const float* xr = x + (long)t * H_DIM;

    float acc = 0.f, sg = 0.f;
    for (int h0 = 0; h0 < H_DIM; h0 += 32) {
        float xv = xr[h0 + lane];
        sg = fmaf(xv, wshg[h0 + lane], sg);
#pragma unroll
        for (int j = 0; j < 32; ++j) {
            float xb = __shfl(xv, j, 32);
            acc = fmaf(xb, gw[(h0 + j) * E_NUM + lane], acc);
        }
    }
    logits[t * E_NUM + lane] = acc;

    // shared-gate sigmoid
    float s = sg;
    for (int m = 16; m >= 1; m >>= 1) s += __shfl_xor(s, m, 32);
    if (lane == 0) sgate[t] = 1.f / (1.f + expf(-s));

    // softmax over 32 lanes
    float mx = acc;
    for (int m = 16; m >= 1; m >>= 1) mx = fmaxf(mx, __shfl_xor(mx, m, 32));
    float p = expf(acc - mx);
    float ssum = p;
    for (int m = 16; m >= 1; m >>= 1) ssum += __shfl_xor(ssum, m, 32);
    float prob = p / ssum;

    // iterative top-4 (ties -> lower index, like jax.lax.top_k)
    float pv = prob;
    for (int k = 0; k < 4; ++k) {
        float v = pv;
        int ii = lane;
        for (int m = 16; m >= 1; m >>= 1) {
            float ov = __shfl_xor(v, m, 32);
            int oi = __shfl_xor(ii, m, 32);
            if (ov > v || (ov == v && oi < ii)) { v = ov; ii = oi; }
        }
        if (lane == 0) { topw[t * 4 + k] = v; sel[t * 4 + k] = ii; }
        if (lane == ii) pv = -1.f;
    }
}

// ---------------------------------------------------------------------------
// K2: capacity-based dispatch (stable order = flat (t,k) order, as reference)
// ---------------------------------------------------------------------------
__global__ void dispatch_kernel(const int* __restrict__ sel,
                                const float* __restrict__ topw,
                                int* __restrict__ idx, float* __restrict__ dw,
                                int* __restrict__ slot, int* __restrict__ cnt_out) {
    for (int i = threadIdx.x; i < E_NUM * CAP; i += 256) idx[i] = -1;
    __syncthreads();
    if (threadIdx.x == 0) {
        int counts[E_NUM];
        for (int e = 0; e < E_NUM; ++e) counts[e] = 0;
        for (int i = 0; i < T_TOK * 4; ++i) {
            int e = sel[i];
            int c = counts[e]++;
            if (c < CAP) {
                idx[e * CAP + c] = i >> 2;
                dw[e * CAP + c] = topw[i];
                slot[i] = c;
            } else {
                slot[i] = -1;
            }
        }
        for (int e = 0; e < E_NUM; ++e) cnt_out[e] = counts[e];
    }
}

// ---------------------------------------------------------------------------
// K3/K4: fused up-projection  T = silu(A@Wg) * (A@Wu). A rows optionally
// gathered via idx (idx==-1 -> zero row). fp32 WMMA 16x16x4; double-buffered
// async global->LDS staging (ASYNCcnt) overlaps HBM with WMMA.
// ---------------------------------------------------------------------------
__global__ __launch_bounds__(256)
void up_swiglu_kernel(const float* __restrict__ X,
                      const float* __restrict__ Wg0,
                      const float* __restrict__ Wu0,
                      const int* __restrict__ idx0,   // nullable
                      const int* __restrict__ cnt0,   // nullable (valid rows / expert)
                      float* __restrict__ Tout0,
                      int N, int K, long wStride, long oStride) {
    __shared__ float sA[2][BM * LDA];
    __shared__ float sBg[2][BK * LDB];
    __shared__ float sBu[2][BK * LDB];

    const int e = blockIdx.z;
    const int nb = blockIdx.x * BN;
    const int mb = blockIdx.y * BM;
    if (cnt0 && cnt0[e] <= mb) return;   // block-uniform: whole tile is padding

    const float* Wg = Wg0 + (long)e * wStride;
    const float* Wu = Wu0 + (long)e * wStride;
    float* Tout = Tout0 + (long)e * oStride;
    const int* idx = idx0 ? (idx0 + e * CAP) : (const int*)0;

    const int tid = threadIdx.x;
    const int lane = tid & 31;
    const int wv = tid >> 5;
    const int wm = (wv & 3) * 32;
    const int wn = (wv >> 2) * 32;
    const int l16 = lane & 15;
    const int hi = lane >> 4;

    const int ar = tid >> 3;       // A staging: row 0..31 (+i*32)
    const int ac = (tid & 7) * 4;  // col (float4)
    const int br = tid >> 4;       // B staging: row 0..15 (+i*16)
    const int bc = (tid & 15) * 4;

    const float* aSrc[4];
#pragma unroll
    for (int i = 0; i < 4; ++i) {
        int grow = mb + ar + i * 32;
        if (idx) {
            int tk = idx[grow];
            aSrc[i] = (tk >= 0) ? (X + (long)tk * K) : (const float*)0;
        } else {
            aSrc[i] = X + (long)grow * K;
        }
    }

    auto stage = [&](int buf, int k0) {
#pragma unroll
        for (int i = 0; i < 4; ++i) {
            int r = ar + i * 32;
            float* dst = &sA[buf][r * LDA + ac];
            if (aSrc[i]) async_b128(dst, aSrc[i] + k0 + ac);
            else *(float4*)dst = make_float4(0.f, 0.f, 0.f, 0.f);
        }
#pragma unroll
        for (int i = 0; i < 2; ++i) {
            int r = br + i * 16;
            async_b128(&sBg[buf][r * LDB + bc], Wg + (long)(k0 + r) * N + nb + bc);
            async_b128(&sBu[buf][r * LDB + bc], Wu + (long)(k0 + r) * N + nb + bc);
        }
    };

    v8f accg[2][2], accu[2][2];
#pragma unroll
    for (int i = 0; i < 2; ++i)
#pragma unroll
        for (int j = 0; j < 2; ++j) { accg[i][j] = zero8(); accu[i][j] = zero8(); }

    stage(0, 0);
    const int S = K / BK;
    for (int s = 0; s < S; ++s) {
        wait_async0();
        __syncthreads();                       // buf (s&1) ready for all waves
        if (s + 1 < S) stage((s + 1) & 1, (s + 1) * BK);
        const float* cA = sA[s & 1];
        const float* cBg = sBg[s & 1];
        const float* cBu = sBu[s & 1];
#pragma unroll
        for (int ks = 0; ks < BK / 4; ++ks) {
            int k = ks * 4 + hi * 2;
            v2f a0 = *(const v2f*)&cA[(wm + l16) * LDA + k];
            v2f a1 = *(const v2f*)&cA[(wm + 16 + l16) * LDA + k];
            v2f bg0 = { cBg[k * LDB + wn + l16],      cBg[(k + 1) * LDB + wn + l16] };
            v2f bg1 = { cBg[k * LDB + wn + 16 + l16], cBg[(k + 1) * LDB + wn + 16 + l16] };
            v2f bu0 = { cBu[k * LDB + wn + l16],      cBu[(k + 1) * LDB + wn + l16] };
            v2f bu1 = { cBu[k * LDB + wn + 16 + l16], cBu[(k + 1) * LDB + wn + 16 + l16] };
            accg[0][0] = __builtin_amdgcn_wmma_f32_16x16x4_f32(false, a0, false, bg0, (short)0, accg[0][0], false, false);
            accg[0][1] = __builtin_amdgcn_wmma_f32_16x16x4_f32(false, a0, false, bg1, (short)0, accg[0][1], false, false);
            accg[1][0] = __builtin_amdgcn_wmma_f32_16x16x4_f32(false, a1, false, bg0, (short)0, accg[1][0], false, false);
            accg[1][1] = __builtin_amdgcn_wmma_f32_16x16x4_f32(false, a1, false, bg1, (short)0, accg[1][1], false, false);
            accu[0][0] = __builtin_amdgcn_wmma_f32_16x16x4_f32(false, a0, false, bu0, (short)0, accu[0][0], false, false);
            accu[0][1] = __builtin_amdgcn_wmma_f32_16x16x4_f32(false, a0, false, bu1, (short)0, accu[0][1], false, false);
            accu[1][0] = __builtin_amdgcn_wmma_f32_16x16x4_f32(false, a1, false, bu0, (short)0, accu[1][0], false, false);
            accu[1][1] = __builtin_amdgcn_wmma_f32_16x16x4_f32(false, a1, false, bu1, (short)0, accu[1][1], false, false);
        }
        __syncthreads();                       // all waves done reading buf (s&1)
    }

#pragma unroll
    for (int ms = 0; ms < 2; ++ms)
#pragma unroll
        for (int ns = 0; ns < 2; ++ns)
#pragma unroll
            for (int v = 0; v < 8; ++v) {
                int row = mb + wm + ms * 16 + v + hi * 8;
                int col = nb + wn + ns * 16 + l16;
                float g = accg[ms][ns][v];
                float u = accu[ms][ns][v];
                float sl = g / (1.f + expf(-g));
                Tout[(long)row * N + col] = sl * u;
            }
}

// ---------------------------------------------------------------------------
// K5/K6: down-projection  Out = (Tin @ Wd) * rowscale(dw). fp32 WMMA with
// double-buffered async LDS staging.
// ---------------------------------------------------------------------------
__global__ __launch_bounds__(256)
void down_kernel(const float* __restrict__ Tin0,
                 const float* __restrict__ Wd0,
                 const float* __restrict__ dw0,   // nullable
                 const int* __restrict__ cnt0,    // nullable
                 float* __restrict__ Out0,
                 int N, int K, long tStride, long wStride, long oStride) {
    __shared__ float sA[2][BM * LDA];
    __shared__ float sB[2][BK * LDB];

    const int e = blockIdx.z;
    const int nb = blockIdx.x * BN;
    const int mb = blockIdx.y * BM;
    if (cnt0 && cnt0[e] <= mb) return;   // skipped consistently with up kernel

    const float* Tin = Tin0 + (long)e * tStride;
    const float* Wd = Wd0 + (long)e * wStride;
    float* Out = Out0 + (long)e * oStride;
    const float* dw = dw0 ? (dw0 + e * CAP) : (const float*)0;

    const int tid = threadIdx.x;
    const int lane = tid & 31;
    const int wv = tid >> 5;
    const int wm = (wv & 3) * 32;
    const int wn = (wv >> 2) * 32;
    const int l16 = lane & 15;
    const int hi = lane >> 4;

    const int ar = tid >> 3;
    const int ac = (tid & 7) * 4;
    const int br = tid >> 4;
    const int bc = (tid & 15) * 4;

    auto stage = [&](int buf, int k0) {
#pragma unroll
        for (int i = 0; i < 4; ++i) {
            int r = ar + i * 32;
            async_b128(&sA[buf][r * LDA + ac], Tin + (long)(mb + r) * K + k0 + ac);
        }
#pragma unroll
        for (int i = 0; i < 2; ++i) {
            int r = br + i * 16;
            async_b128(&sB[buf][r * LDB + bc], Wd + (long)(k0 + r) * N + nb + bc);
        }
    };

    v8f acc[2][2];
#pragma unroll
    for (int i = 0; i < 2; ++i)
#pragma unroll
        for (int j = 0; j < 2; ++j) acc[i][j] = zero8();

    stage(0, 0);
    const int S = K / BK;
    for (int s = 0; s < S; ++s) {
        wait_async0();
        __syncthreads();
        if (s + 1 < S) stage((s + 1) & 1, (s + 1) * BK);
        const float* cA = sA[s & 1];
        const float* cB = sB[s & 1];
#pragma unroll
        for (int ks = 0; ks < BK / 4; ++ks) {
            int k = ks * 4 + hi * 2;
            v2f a0 = *(const v2f*)&cA[(wm + l16) * LDA + k];
            v2f a1 = *(const v2f*)&cA[(wm + 16 + l16) * LDA + k];
            v2f b0 = { cB[k * LDB + wn + l16],      cB[(k + 1) * LDB + wn + l16] };
            v2f b1 = { cB[k * LDB + wn + 16 + l16], cB[(k + 1) * LDB + wn + 16 + l16] };
            acc[0][0] = __builtin_amdgcn_wmma_f32_16x16x4_f32(false, a0, false, b0, (short)0, acc[0][0], false, false);
            acc[0][1] = __builtin_amdgcn_wmma_f32_16x16x4_f32(false, a0, false, b1, (short)0, acc[0][1], false, false);
            acc[1][0] = __builtin_amdgcn_wmma_f32_16x16x4_f32(false, a1, false, b0, (short)0, acc[1][0], false, false);
            acc[1][1] = __builtin_amdgcn_wmma_f32_16x16x4_f32(false, a1, false, b1, (short)0, acc[1][1], false, false);
        }
        __syncthreads();
    }

#pragma unroll
    for (int ms = 0; ms < 2; ++ms)
#pragma unroll
        for (int ns = 0; ns < 2; ++ns)
#pragma unroll
            for (int v = 0; v < 8; ++v) {
                int row = mb + wm + ms * 16 + v + hi * 8;
                int col = nb + wn + ns * 16 + l16;
                float sc = dw ? dw[row] : 1.0f;
                Out[(long)row * N + col] = acc[ms][ns][v] * sc;
            }
}

// ---------------------------------------------------------------------------
// K7: final combine  out[t,n] = sgate[t]*shared[t,n] + sum_k h[e_k, slot_k, n]
// ---------------------------------------------------------------------------
__global__ __launch_bounds__(256)
void combine_kernel(const float* __restrict__ shr, const float* __restrict__ h,
                    const float* __restrict__ sgate, const int* __restrict__ sel,
                    const int* __restrict__ slot, float* __restrict__ out) {
    int gid = blockIdx.x * 256 + threadIdx.x;
    int t = gid >> 11;      // / H_DIM
    int n = gid & 2047;
    float v = sgate[t] * shr[gid];
#pragma unroll
    for (int k = 0; k < 4; ++k) {
        int s = slot[t * 4 + k];
        if (s >= 0) {
            int e = sel[t * 4 + k];
            v += h[((long)e * CAP + s) * H_DIM + n];
        }
    }
    out[gid] = v;
}

extern "C" void kernel_launch(void* const* d_in, const int* in_sizes, int n_in,
                              void* d_out, int out_size, void* d_ws, size_t ws_size,
                              hipStream_t stream) {
    const float* x    = (const float*)d_in[0];
    const float* gw   = (const float*)d_in[1];
    const float* wg   = (const float*)d_in[2];
    const float* wu   = (const float*)d_in[3];
    const float* wd   = (const float*)d_in[4];
    const float* wsg  = (const float*)d_in[5];
    const float* wsu  = (const float*)d_in[6];
    const float* wsd  = (const float*)d_in[7];
    const float* wshg = (const float*)d_in[8];

    float* out = (float*)d_out;
    float* final_out = out;                                  // [1024*2048]
    float* logits_out = out + (long)T_TOK * H_DIM;           // [1024*32]

    float* wsf   = (float*)d_ws;
    float* topw  = wsf;
    int*   sel   = (int*)(wsf + 4096);
    int*   slot  = (int*)(wsf + 8192);
    float* sgate = wsf + 12288;
    int*   idx   = (int*)(wsf + 13312);
    float* dw    = wsf + 21504;
    int*   cnt   = (int*)(wsf + 29696);                      // [32]
    float* t_moe = wsf + 29728;                              // [32*256*1408]
    float* ts    = t_moe + (long)E_NUM * CAP * I_DIM;        // [1024*5632]
    float* shr   = ts + (long)T_TOK * IS_DIM;                // [1024*2048]
    float* h_moe = shr + (long)T_TOK * H_DIM;                // [32*256*2048]

    router_kernel<<<T_TOK / 8, 256, 0, stream>>>(x, gw, wshg, logits_out, topw, sel, sgate);
    dispatch_kernel<<<1, 256, 0, stream>>>(sel, topw, idx, dw, slot, cnt);

    // Expert up-proj (gathered), N=1408, K=2048
    up_swiglu_kernel<<<dim3(I_DIM / BN, CAP / BM, E_NUM), 256, 0, stream>>>(
        x, wg, wu, idx, cnt, t_moe, I_DIM, H_DIM, (long)H_DIM * I_DIM, (long)CAP * I_DIM);
    // Shared up-proj, N=5632, K=2048
    up_swiglu_kernel<<<dim3(IS_DIM / BN, T_TOK / BM, 1), 256, 0, stream>>>(
        x, wsg, wsu, (const int*)0, (const int*)0, ts, IS_DIM, H_DIM, 0L, 0L);
    // Shared down-proj, N=2048, K=5632
    down_kernel<<<dim3(H_DIM / BN, T_TOK / BM, 1), 256, 0, stream>>>(
        ts, wsd, (const float*)0, (const int*)0, shr, H_DIM, IS_DIM, 0L, 0L, 0L);
    // Expert down-proj with dw row-scale, N=2048, K=1408
    down_kernel<<<dim3(H_DIM / BN, CAP / BM, E_NUM), 256, 0, stream>>>(
        t_moe, wd, dw, cnt, h_moe, H_DIM, I_DIM, (long)CAP * I_DIM,
        (long)I_DIM * H_DIM, (long)CAP * H_DIM);

    combine_kernel<<<(T_TOK * H_DIM) / 256, 256, 0, stream>>>(
        shr, h_moe, sgate, sel, slot, final_out);
}